// PhysicsPrior_17523466568033
// MI455X (gfx1250) — compile-verified
//
#include <hip/hip_runtime.h>
#include <cstdint>

#ifndef __has_builtin
#define __has_builtin(x) 0
#endif

#if __has_builtin(__builtin_amdgcn_global_load_async_to_lds_b32) && \
    __has_builtin(__builtin_amdgcn_s_wait_asynccnt)
#define USE_ASYNC_LDS 1
#else
#define USE_ASYNC_LDS 0
#endif

#define TW 32          // tile width  (output pixels)
#define TTH 16         // tile height (output pixels)
#define HAW 40         // haloed width  (TW + 8)
#define HAH 24         // haloed height (TTH + 8)
#define HALO_N (HAW * HAH)   // 960 floats = 3.84 KB LDS
#define NTHREADS 256
#define LOG2E 1.44269504088896340736f

#define IMG_H 256
#define IMG_W 256

typedef __attribute__((address_space(1))) int  g_int;
typedef __attribute__((address_space(3))) int  l_int;

__device__ __forceinline__ float fast_exp2(float x) {
#if __has_builtin(__builtin_amdgcn_exp2f)
  return __builtin_amdgcn_exp2f(x);   // v_exp_f32
#else
  return exp2f(x);
#endif
}

// ---- compile-time direction table -----------------------------------------
constexpr int igcd(int a, int b) {
  a = a < 0 ? -a : a;
  b = b < 0 ? -b : b;
  while (b) { int t = a % b; a = b; b = t; }
  return a;
}

#define NDIR 49
struct DirTab {
  int px[NDIR];   // primitive direction x
  int py[NDIR];   // primitive direction y
  int id[81];     // tap (row-major) -> direction index
  int n;
};

constexpr DirTab make_dirtab() {
  DirTab t{};
  t.n = 0;
  for (int dy = -4; dy <= 4; ++dy)
    for (int dx = -4; dx <= 4; ++dx) {
      bool prim = (dx == 0 && dy == 0) || (igcd(dx, dy) == 1);
      if (prim) { t.px[t.n] = dx; t.py[t.n] = dy; ++t.n; }
    }
  for (int dy = -4; dy <= 4; ++dy)
    for (int dx = -4; dx <= 4; ++dx) {
      int g0 = igcd(dx, dy);
      int g  = (g0 == 0) ? 1 : g0;
      int px = dx / g, py = dy / g;
      for (int d = 0; d < t.n; ++d)
        if (t.px[d] == px && t.py[d] == py)
          t.id[(dy + 4) * 9 + (dx + 4)] = d;
    }
  return t;
}

constexpr DirTab DT = make_dirtab();
static_assert(make_dirtab().n == NDIR, "expected 49 primitive directions");
// ---------------------------------------------------------------------------

__global__ __launch_bounds__(NTHREADS)
void PhysicsPrior_kernel(const float* __restrict__ fire,
                         const float* __restrict__ u,
                         const float* __restrict__ v,
                         const float* __restrict__ slope,
                         const float* __restrict__ aspect,
                         const float* __restrict__ temp,
                         const float* __restrict__ rh,
                         const float* __restrict__ ndvi,
                         float* __restrict__ out) {
  __shared__ float s_fire[HALO_N];

  const int b   = blockIdx.z;
  const int x0  = blockIdx.x * TW;
  const int y0  = blockIdx.y * TTH;
  const int tid = threadIdx.x;

  const float* fire_b = fire + (size_t)b * IMG_H * IMG_W;

  // ---- Stage haloed prev_fire tile into LDS (zero padding outside image) ----
  #pragma unroll
  for (int it = 0; it < (HALO_N + NTHREADS - 1) / NTHREADS; ++it) {
    int e = tid + it * NTHREADS;
    if (e < HALO_N) {
      int ly = e / HAW;
      int lx = e - ly * HAW;
      int gy = y0 - 4 + ly;
      int gx = x0 - 4 + lx;
      bool inb = (gy >= 0) & (gy < IMG_H) & (gx >= 0) & (gx < IMG_W);
#if USE_ASYNC_LDS
      if (inb) {
        // DMA copy global -> LDS, tracked with ASYNCcnt (CDNA5 async path).
        __builtin_amdgcn_global_load_async_to_lds_b32(
            (g_int*)(uintptr_t)(fire_b + gy * IMG_W + gx),
            (l_int*)(uint32_t)(uintptr_t)(&s_fire[e]),
            0, 0);
      } else {
        s_fire[e] = 0.0f;            // ds_store zero (disjoint words, no race)
      }
#else
      s_fire[e] = inb ? fire_b[gy * IMG_W + gx] : 0.0f;
#endif
    }
  }
#if USE_ASYNC_LDS
  __builtin_amdgcn_s_wait_asynccnt(0);
#endif
  __syncthreads();

  // ---- Per-pixel adaptive 81-tap kernel ----
  // ker[k] = exp2( ca_k*X + sa_k*Y ); (ca_k,sa_k) is the tap's unit direction,
  // so collinear taps share weights: 49 distinct directions among 81 taps.
  // Phase 1: 49 independent fma+exp chains (good ILP, SROA-promoted w[]).
  // Phase 2: row-major taps -> sequential LDS offsets (ds_load_2addr pairs).
  #pragma unroll
  for (int pp = 0; pp < (TW * TTH) / NTHREADS; ++pp) {
    const int p  = tid + pp * NTHREADS;
    const int ty = p >> 5;          // / TW
    const int tx = p & 31;          // % TW
    const int gy = y0 + ty;
    const int gx = x0 + tx;
    const int idx = (b * IMG_H + gy) * IMG_W + gx;

    const float uu = u[idx];
    const float vv = v[idx];
    const float fx = uu + 1e-8f;
    const float r  = sqrtf(fx * fx + vv * vv);
    const float ir = (r > 0.0f) ? (1.0f / r) : 0.0f;
    const float cw = (r > 0.0f) ? (fx * ir) : 1.0f;   // cos(wind_angle)
    const float sw = vv * ir;                          // sin(wind_angle)
    const float ws = fminf(sqrtf(uu * uu + vv * vv) * 0.1f, 1.0f);

    const float sn  = fminf(fmaxf(slope[idx], 0.0f), 1.0f);
    const float asp = aspect[idx];
    float sA, cA;
    sincosf(asp, &sA, &cA);

    const float X = (ws * cw + sn * cA) * LOG2E;
    const float Y = (ws * sw + sn * sA) * LOG2E;

    // Phase 1: one v_exp_f32 per distinct direction (unconditional writes,
    // constant indices -> promoted to registers).
    float w[NDIR];
    #pragma unroll
    for (int d = 0; d < NDIR; ++d) {
      const float fdx = (float)DT.px[d] + 1e-8f;
      const float fdy = (float)DT.py[d];
      const float rr  = sqrtf(fdx * fdx + fdy * fdy);   // folds to literal
      const float ca  = fdx / rr;
      const float sa  = fdy / rr;
      w[d] = fast_exp2(ca * X + sa * Y);                // v_fma + v_exp_f32
    }

    // Phase 2: row-major 81-tap accumulation (sequential LDS offsets).
    const float* base = &s_fire[ty * HAW + tx];
    float wsum = 0.0f;
    float acc  = 0.0f;
    #pragma unroll
    for (int k = 0; k < 81; ++k) {
      const float ww = w[DT.id[k]];                     // constant index
      wsum += ww;
      acc  += ww * base[(k / 9) * HAW + (k % 9)];       // ds_load + v_fmac
    }

    const float z    = 0.03f * temp[idx] - 0.02f * rh[idx] + 0.7f * ndvi[idx];
    const float damp = 1.0f / (1.0f + fast_exp2(-z * LOG2E));
    const float spread = (acc / (wsum + 1e-8f)) * damp;

    const size_t hw  = (size_t)IMG_H * IMG_W;
    const size_t pix = (size_t)gy * IMG_W + gx;
    float* ob = out + (size_t)b * 4 * hw;
    ob[0 * hw + pix] = spread;
    ob[1 * hw + pix] = cw;
    ob[2 * hw + pix] = sw;
    ob[3 * hw + pix] = damp;
  }
}

extern "C" void kernel_launch(void* const* d_in, const int* in_sizes, int n_in,
                              void* d_out, int out_size, void* d_ws, size_t ws_size,
                              hipStream_t stream) {
  const float* fire   = (const float*)d_in[0];
  const float* u      = (const float*)d_in[1];
  const float* v      = (const float*)d_in[2];
  const float* slope  = (const float*)d_in[3];
  const float* aspect = (const float*)d_in[4];
  const float* temp   = (const float*)d_in[5];
  const float* rh     = (const float*)d_in[6];
  const float* ndvi   = (const float*)d_in[7];
  float* out = (float*)d_out;

  const int B = in_sizes[0] / (IMG_H * IMG_W);   // = 8
  dim3 grid(IMG_W / TW, IMG_H / TTH, B);         // 8 x 16 x 8 = 1024 blocks
  PhysicsPrior_kernel<<<grid, NTHREADS, 0, stream>>>(
      fire, u, v, slope, aspect, temp, rh, ndvi, out);
}